// SRGNN_25692494365145
// MI455X (gfx1250) — compile-verified
//
#include <hip/hip_runtime.h>
#include <hip/hip_bf16.h>
#include <cstdint>

#define NEG_SLOPE 0.01f
#define DFEAT 128   // reference D

__device__ __forceinline__ float lrelu(float v) { return v > 0.f ? v : NEG_SLOPE * v; }

// monotonic float <-> uint encoding for atomicMax-based segment max
__device__ __forceinline__ unsigned enc_f(float f) {
    unsigned u = __float_as_uint(f);
    return (u & 0x80000000u) ? ~u : (u | 0x80000000u);
}
__device__ __forceinline__ float dec_f(unsigned u) {
    unsigned b = (u & 0x80000000u) ? (u & 0x7fffffffu) : ~u;
    return __uint_as_float(b);
}

// exp_dst(e) for the reference's concatenated layout
__device__ __forceinline__ int exp_dst_of(const int* __restrict__ ei, int e, int E) {
    return (e < E) ? ei[E + e] : ei[e - E];
}

// ---------------- kernel 0: init out accumulator, segmax, denom ----------------
__global__ void k_init(float* __restrict__ out, unsigned* __restrict__ segmax,
                       float* __restrict__ denom, int N, int total4) {
    int idx = blockIdx.x * blockDim.x + threadIdx.x;
    if (idx < total4) {
        float4 z = make_float4(0.f, 0.f, 0.f, 0.f);
        ((float4*)out)[idx] = z;
    }
    if (idx < N) {
        segmax[idx] = 0x007FFFFFu;   // enc_f(-inf)
        denom[idx]  = 0.f;
    }
}

// ---------------- kernel 1: per-edge logits + segment max (wave32 per edge) ----------------
__global__ __launch_bounds__(256)
void k_logits(const float* __restrict__ x, const float* __restrict__ Wa,
              const float* __restrict__ Wf, const float* __restrict__ Wb,
              const float* __restrict__ lsa, const int* __restrict__ ei,
              const int* __restrict__ batch,
              float* __restrict__ logits, unsigned* __restrict__ segmax,
              int E, int EXP) {
    int wave = (blockIdx.x * blockDim.x + threadIdx.x) >> 5;
    int lane = threadIdx.x & 31;
    if (wave >= EXP) return;
    int e = wave;
    int s = ei[e];                       // exp_src(e) == ei[e] for all e
    int d = exp_dst_of(ei, e, E);
    int bq = batch[d];

    const float4* xi = (const float4*)(x   + (size_t)d  * DFEAT);
    const float4* xj = (const float4*)(x   + (size_t)s  * DFEAT);
    const float4* qv = (const float4*)(lsa + (size_t)bq * DFEAT);
    const float4* wa = (const float4*)Wa;
    const float4* ws = (const float4*)((e < E) ? Wf : Wb);

    float4 vi = xi[lane], vj = xj[lane], vq = qv[lane], va = wa[lane], vs = ws[lane];

    float pa = vi.x * vq.x * va.x + vi.y * vq.y * va.y +
               vi.z * vq.z * va.z + vi.w * vq.w * va.w;
    float pb = vi.x * vj.x * vs.x + vi.y * vj.y * vs.y +
               vi.z * vj.z * vs.z + vi.w * vj.w * vs.w;

#pragma unroll
    for (int off = 16; off > 0; off >>= 1) {
        pa += __shfl_xor(pa, off, 32);
        pb += __shfl_xor(pb, off, 32);
    }
    if (lane == 0) {
        float lg = lrelu(pb) + lrelu(pa);
        logits[e] = lg;
        atomicMax(segmax + d, enc_f(lg));
    }
}

// ---------------- kernel 2: exp(logit - max) and denom segment sum ----------------
__global__ void k_ex(float* __restrict__ exbuf, const unsigned* __restrict__ segmax,
                     float* __restrict__ denom, const int* __restrict__ ei,
                     int E, int EXP) {
    int e = blockIdx.x * blockDim.x + threadIdx.x;
    if (e >= EXP) return;
    int d = exp_dst_of(ei, e, E);
    float ex = __expf(exbuf[e] - dec_f(segmax[d]));
    exbuf[e] = ex;
    atomicAdd(denom + d, ex);
}

// ---------------- kernel 3: weighted scatter-add with async global->LDS staging ----------------
__global__ __launch_bounds__(256)
void k_aggr(const float* __restrict__ x, const int* __restrict__ ei,
            const float* __restrict__ exbuf, const float* __restrict__ denom,
            float* __restrict__ out, int E, int EXP) {
    __shared__ float smem[8][2][DFEAT];   // 8 waves/block x double buffer x 512B row
    int lane  = threadIdx.x & 31;
    int wslot = threadIdx.x >> 5;
    int gwave = (blockIdx.x * blockDim.x + threadIdx.x) >> 5;
    int nwav  = (gridDim.x * blockDim.x) >> 5;

    // LDS byte addresses (low 32 bits of the flat shared address are the LDS offset)
    unsigned lds0 = (unsigned)(uintptr_t)&smem[wslot][0][lane * 4];
    unsigned lds1 = (unsigned)(uintptr_t)&smem[wslot][1][lane * 4];

    int e = gwave;
    if (e < EXP) {
        int s = ei[e];
        uint64_t ga = (uint64_t)(uintptr_t)(x + (size_t)s * DFEAT + lane * 4);
        asm volatile("global_load_async_to_lds_b128 %0, %1, off"
                     :: "v"(lds0), "v"(ga) : "memory");
    }
    int cur = 0;
    for (; e < EXP; e += nwav) {
        int enext = e + nwav;
        if (enext < EXP) {
            int s2 = ei[enext];
            uint64_t ga2 = (uint64_t)(uintptr_t)(x + (size_t)s2 * DFEAT + lane * 4);
            unsigned ldsn = cur ? lds0 : lds1;
            asm volatile("global_load_async_to_lds_b128 %0, %1, off"
                         :: "v"(ldsn), "v"(ga2) : "memory");
            asm volatile("s_wait_asynccnt 0x1" ::: "memory");  // oldest fetch done
        } else {
            asm volatile("s_wait_asynccnt 0x0" ::: "memory");
        }
        int d = exp_dst_of(ei, e, E);
        float w = exbuf[e] / denom[d];
        const float* row = &smem[wslot][cur][lane * 4];
        float* op = out + (size_t)d * DFEAT + lane * 4;
        atomicAdd(op + 0, w * row[0]);
        atomicAdd(op + 1, w * row[1]);
        atomicAdd(op + 2, w * row[2]);
        atomicAdd(op + 3, w * row[3]);
        cur ^= 1;
    }
}

// ---------------- kernel 4: out = mask ? aggr + x : 2x ----------------
__global__ void k_final(float* __restrict__ out, const float* __restrict__ x,
                        const int* __restrict__ mask, int total4) {
    int idx = blockIdx.x * blockDim.x + threadIdx.x;
    if (idx >= total4) return;
    int n = idx >> 5;                    // 32 float4 per 128-wide row
    float4 a  = ((const float4*)out)[idx];
    float4 xv = ((const float4*)x)[idx];
    float4 r;
    if (mask[n] == 1) {
        r.x = a.x + xv.x; r.y = a.y + xv.y; r.z = a.z + xv.z; r.w = a.w + xv.w;
    } else {
        r.x = xv.x + xv.x; r.y = xv.y + xv.y; r.z = xv.z + xv.z; r.w = xv.w + xv.w;
    }
    ((float4*)out)[idx] = r;
}

extern "C" void kernel_launch(void* const* d_in, const int* in_sizes, int n_in,
                              void* d_out, int out_size, void* d_ws, size_t ws_size,
                              hipStream_t stream) {
    const float* x    = (const float*)d_in[0];
    const float* Wa   = (const float*)d_in[1];
    const float* Wf   = (const float*)d_in[2];
    const float* Wb   = (const float*)d_in[3];
    const float* lsa  = (const float*)d_in[4];
    const int*   ei   = (const int*)d_in[5];
    const int*   bat  = (const int*)d_in[6];
    const int*   mask = (const int*)d_in[7];
    // d_in[8] = n_nonself scalar on device; derive host-side from sizes instead.

    const int E   = in_sizes[5] / 2;   // 300000 (src||loops rows)
    const int N   = in_sizes[6];       // 50000
    const int ENS = E - N;             // 250000 non-self edges
    const int EXP = E + ENS;           // 550000 expanded edges

    float* out = (float*)d_out;

    // workspace layout (aligned to 256B chunks)
    char* ws = (char*)d_ws;
    float*    exbuf  = (float*)ws;
    size_t    o1     = ((size_t)EXP * sizeof(float) + 255) & ~(size_t)255;
    unsigned* segmax = (unsigned*)(ws + o1);
    size_t    o2     = o1 + (((size_t)N * sizeof(unsigned) + 255) & ~(size_t)255);
    float*    denom  = (float*)(ws + o2);

    const int total4 = N * (DFEAT / 4);

    k_init<<<(total4 + 255) / 256, 256, 0, stream>>>(out, segmax, denom, N, total4);

    {   // one wave32 per edge
        long long threads = (long long)EXP * 32;
        int blocks = (int)((threads + 255) / 256);
        k_logits<<<blocks, 256, 0, stream>>>(x, Wa, Wf, Wb, lsa, ei, bat,
                                             exbuf, segmax, E, EXP);
    }

    k_ex<<<(EXP + 255) / 256, 256, 0, stream>>>(exbuf, segmax, denom, ei, E, EXP);

    {   // ~131072 waves -> ~4 edges per wave, enabling async double-buffering
        int blocks = 16384;
        k_aggr<<<blocks, 256, 0, stream>>>(x, ei, exbuf, denom, out, E, EXP);
    }

    k_final<<<(total4 + 255) / 256, 256, 0, stream>>>(out, x, mask, total4);
}